// OutputLayer_21191368639071
// MI455X (gfx1250) — compile-verified
//
#include <hip/hip_runtime.h>

#define N_NODES 50000
#define N_EDGES 500000
#define EMBED   128
#define RBF     16
#define N_OUT   12
#define N_DENSE 3
#define XS      132   // padded LDS row stride (floats) to avoid bank conflicts

typedef __attribute__((ext_vector_type(2))) float v2f;
typedef __attribute__((ext_vector_type(8))) float v8f;

__device__ __forceinline__ float ssp_act(float x) {
    // shifted softplus: softplus(x) - ln(2), numerically stable
    float sp = fmaxf(x, 0.0f) + log1pf(__expf(-fabsf(x)));
    return sp - 0.69314718055994530942f;
}

// ---------------------------------------------------------------------------
// Kernel 0: zero the atom accumulator in workspace
// ---------------------------------------------------------------------------
__global__ void zero_kernel(float* __restrict__ p, int n) {
    int i = blockIdx.x * blockDim.x + threadIdx.x;
    if (i < n) p[i] = 0.0f;
}

// ---------------------------------------------------------------------------
// Kernel 1: e_ji = rbf_ji @ lin_rbf_w^T ; msg = e_ji * m_ji ; scatter-add.
// One wave (32 lanes) per edge, 4 embed dims per lane. lin_rbf_w (8 KB)
// cached in registers (4 rows x 16 per lane). m_ji streamed as float4
// (fully coalesced); atomics land in L2 (atom_i = 25.6 MB << 192 MB L2).
// ---------------------------------------------------------------------------
__global__ void __launch_bounds__(256)
edge_message_scatter(const float* __restrict__ m_ji,
                     const float* __restrict__ rbf_ji,
                     const int*   __restrict__ edge_index,
                     const float* __restrict__ lin_rbf_w,
                     float*       __restrict__ atom)
{
    const int lane          = threadIdx.x & 31;
    const int waveInBlock   = threadIdx.x >> 5;
    const int wavesPerBlock = blockDim.x >> 5;
    const int waveId        = blockIdx.x * wavesPerBlock + waveInBlock;
    const int nWaves        = gridDim.x * wavesPerBlock;

    // cache my 4 rows of lin_rbf_w (rows lane*4 .. lane*4+3)
    float w[4][RBF];
    #pragma unroll
    for (int d = 0; d < 4; ++d)
        #pragma unroll
        for (int r = 0; r < RBF; ++r)
            w[d][r] = lin_rbf_w[(lane * 4 + d) * RBF + r];

    for (int e = waveId; e < N_EDGES; e += nWaves) {
        float rb[RBF];
        #pragma unroll
        for (int i = 0; i < RBF; ++i) rb[i] = rbf_ji[e * RBF + i]; // broadcast

        float4 m = *(const float4*)(m_ji + (size_t)e * EMBED + lane * 4);
        int dst = edge_index[N_EDGES + e];   // row 1 = destination atom

        float acc[4];
        #pragma unroll
        for (int d = 0; d < 4; ++d) {
            float s = 0.0f;
            #pragma unroll
            for (int i = 0; i < RBF; ++i) s = fmaf(rb[i], w[d][i], s);
            acc[d] = s;
        }
        acc[0] *= m.x; acc[1] *= m.y; acc[2] *= m.z; acc[3] *= m.w;

        float* ap = atom + (size_t)dst * EMBED + lane * 4;
        atomicAdd(ap + 0, acc[0]);
        atomicAdd(ap + 1, acc[1]);
        atomicAdd(ap + 2, acc[2]);
        atomicAdd(ap + 3, acc[3]);
    }
}

// ---------------------------------------------------------------------------
// Kernel 2: fused 3x dense(128x128)+ssp + output(128->12), WMMA f32 16x16x4.
// Block = 128 threads = 4 waves; each wave owns a 16-row tile of atom_i in
// padded LDS. Per layer: 8 N-tiles x 32 K-steps of v_wmma_f32_16x16x4_f32,
// bias preloaded into the C accumulator, ssp on D, transpose back via LDS.
// ---------------------------------------------------------------------------
__global__ void __launch_bounds__(128)
dense_out_kernel(const float* __restrict__ atom,
                 const float* __restrict__ dense_w,
                 const float* __restrict__ dense_b,
                 const float* __restrict__ out_w,
                 float*       __restrict__ out)
{
    __shared__ float xs[4][16 * XS];   // 4 waves * 8.25 KB = 33 KB

    const int lane = threadIdx.x & 31;
    const int wave = threadIdx.x >> 5;
    const int l15  = lane & 15;        // column within 16-wide tile
    const int half = lane >> 4;        // 0: K pair {0,1} / rows 0-7 ; 1: {2,3} / 8-15
    float* X = xs[wave];
    const int n0 = (blockIdx.x * 4 + wave) * 16;

    // ---- load 16x128 activation tile (guarded, zero-fill past N_NODES) ----
    #pragma unroll 4
    for (int r = 0; r < 16; ++r) {
        int n = n0 + r;
        float4 v;
        if (n < N_NODES) v = *(const float4*)(atom + (size_t)n * EMBED + lane * 4);
        else             { v.x = 0.f; v.y = 0.f; v.z = 0.f; v.w = 0.f; }
        *(float4*)(X + r * XS + lane * 4) = v;
    }
    __syncthreads();

    // ---- three dense layers ----
    for (int layer = 0; layer < N_DENSE; ++layer) {
        const float* W = dense_w + layer * EMBED * EMBED;   // [out e][in d]
        const float* B = dense_b + layer * EMBED;

        v8f acc[8];
        #pragma unroll
        for (int t = 0; t < 8; ++t) {
            float bv = B[t * 16 + l15];                     // bias per column
            acc[t] = (v8f){bv, bv, bv, bv, bv, bv, bv, bv}; // C = bias
        }

        for (int k = 0; k < EMBED; k += 4) {
            int kb = k + half * 2;
            // A fragment: lane holds X[row=l15][kb], X[row=l15][kb+1]
            float2 av = *(const float2*)(X + l15 * XS + kb);
            v2f a; a.x = av.x; a.y = av.y;
            #pragma unroll
            for (int t = 0; t < 8; ++t) {
                int col = t * 16 + l15;
                // B fragment: B[d][e] = W[e][d]  ->  W[col][kb], W[col][kb+1]
                float2 bv2 = *(const float2*)(W + col * EMBED + kb);
                v2f b; b.x = bv2.x; b.y = bv2.y;
                acc[t] = __builtin_amdgcn_wmma_f32_16x16x4_f32(
                    false, a, false, b, (short)0, acc[t], false, false);
            }
        }

        __syncthreads();
        // ssp + write back in row-major for next layer's A fragments
        #pragma unroll
        for (int t = 0; t < 8; ++t)
            #pragma unroll
            for (int v = 0; v < 8; ++v)
                X[(v + half * 8) * XS + t * 16 + l15] = ssp_act(acc[t][v]);
        __syncthreads();
    }

    // ---- output projection: 128 -> 12 (columns 12..15 zero-padded) ----
    v8f acc = (v8f){0.f, 0.f, 0.f, 0.f, 0.f, 0.f, 0.f, 0.f};
    for (int k = 0; k < EMBED; k += 4) {
        int kb = k + half * 2;
        float2 av = *(const float2*)(X + l15 * XS + kb);
        v2f a; a.x = av.x; a.y = av.y;
        v2f b;
        if (l15 < N_OUT) {
            float2 bv2 = *(const float2*)(out_w + l15 * EMBED + kb);
            b.x = bv2.x; b.y = bv2.y;
        } else {
            b.x = 0.f; b.y = 0.f;
        }
        acc = __builtin_amdgcn_wmma_f32_16x16x4_f32(
            false, a, false, b, (short)0, acc, false, false);
    }

    if (l15 < N_OUT) {
        #pragma unroll
        for (int v = 0; v < 8; ++v) {
            int n = n0 + v + half * 8;   // D: VGPR v = row v (lanes 0-15) / v+8 (16-31)
            if (n < N_NODES) out[n * N_OUT + l15] = acc[v];
        }
    }
}

// ---------------------------------------------------------------------------
extern "C" void kernel_launch(void* const* d_in, const int* in_sizes, int n_in,
                              void* d_out, int out_size, void* d_ws, size_t ws_size,
                              hipStream_t stream) {
    const float* m_ji       = (const float*)d_in[0];
    const float* rbf_ji     = (const float*)d_in[1];
    const int*   edge_index = (const int*)  d_in[2];
    const float* lin_rbf_w  = (const float*)d_in[3];
    const float* dense_w    = (const float*)d_in[4];
    const float* dense_b    = (const float*)d_in[5];
    const float* out_w      = (const float*)d_in[6];
    float*       out        = (float*)d_out;
    float*       atom       = (float*)d_ws;   // 50000*128 floats = 25.6 MB scratch

    // 1) zero accumulator
    int n = N_NODES * EMBED;
    zero_kernel<<<(n + 255) / 256, 256, 0, stream>>>(atom, n);

    // 2) edge messages + scatter-add (grid-stride waves; ~15 edges/wave
    //    amortizes the per-wave lin_rbf_w register preload)
    edge_message_scatter<<<4096, 256, 0, stream>>>(m_ji, rbf_ji, edge_index,
                                                   lin_rbf_w, atom);

    // 3) fused dense stack + output projection (WMMA)
    int tiles  = (N_NODES + 15) / 16;        // 3125
    int blocks = (tiles + 3) / 4;            // 782 (4 waves/block, 16 rows/wave)
    dense_out_kernel<<<blocks, 128, 0, stream>>>(atom, dense_w, dense_b, out_w, out);
}